// OT_GNN_layer_35510789604052
// MI455X (gfx1250) — compile-verified
//
#include <hip/hip_runtime.h>
#include <hip/hip_bf16.h>

// ---------------- problem constants (match reference) ----------------
#define N_NODES 20000
#define DEG     12
#define KLOC    13          // DEG + 1 local nodes
#define N_FEAT  128
#define N_TPL   10
#define M_TPL   8
#define NT80    (N_TPL * M_TPL)   // 80 template rows total
#define N_CLS   8
#define EPS_R   0.1f
#define ALPHA_R 0.5f
#define N_OUTER 5
#define N_SINK  5

typedef __attribute__((ext_vector_type(2))) float v2f;
typedef __attribute__((ext_vector_type(8))) float v8f;

// ---------------------------------------------------------------------
// Kernel 0: template precompute  t2[t*8+j] = ||tf[t,j,:]||^2
//                                c2[t*8+i] = (1/8) * sum_j ts[t,i,j]^2
// ---------------------------------------------------------------------
__global__ void k_tpl_pre(const float* __restrict__ tf, const float* __restrict__ ts,
                          float* __restrict__ t2, float* __restrict__ c2) {
    int tid = blockIdx.x * blockDim.x + threadIdx.x;
    if (tid < NT80) {
        const float* row = tf + tid * N_FEAT;
        float s = 0.f;
        for (int f = 0; f < N_FEAT; ++f) s += row[f] * row[f];
        t2[tid] = s;
    } else if (tid < 2 * NT80) {
        int r = tid - NT80;                 // r = t*8 + i
        const float* row = ts + r * M_TPL;
        float s = 0.f;
        for (int j = 0; j < M_TPL; ++j) s += row[j] * row[j];
        c2[r] = s * (1.0f / M_TPL);
    }
}

// ---------------------------------------------------------------------
// Kernel A: per-node squared row norms x2[n] = ||x[n,:]||^2
// ---------------------------------------------------------------------
__global__ void k_x2(const float* __restrict__ x, float* __restrict__ x2) {
    int n = blockIdx.x * blockDim.x + threadIdx.x;
    if (n >= N_NODES) return;
    const float4* r4 = (const float4*)(x + n * N_FEAT);
    float s = 0.f;
    for (int f = 0; f < N_FEAT / 4; ++f) {
        float4 v = r4[f];
        s += v.x * v.x + v.y * v.y + v.z * v.z + v.w * v.w;
    }
    x2[n] = s;
}

// ---------------------------------------------------------------------
// Kernel 1: E = x @ tf_all^T  ([20000x128] x [128x80]) via f32 WMMA.
// One wave per 16x16 output tile, K swept in steps of 4.
//   A (16x4 f32) layout: lane l -> M = l%16, K-pair base = (l/16)*2 (2 VGPRs)
//   B (4x16 f32) layout: mirrored (B row k striped across lanes; tf row is
//   the N index, so each lane loads a contiguous K-pair from its tf row)
//   C/D (16x16 f32): lane l -> N = l%16; VGPR r -> M = r + 8*(l/16)
// ---------------------------------------------------------------------
__global__ void k_gemm_E(const float* __restrict__ x, const float* __restrict__ tf,
                         float* __restrict__ E) {
    const int m0   = blockIdx.x * 16;
    const int n0   = blockIdx.y * 16;
    const int l    = threadIdx.x;        // 0..31
    const int half = l >> 4;             // 0/1 -> K-pair select
    const int lm   = l & 15;

    const float* arow = x  + (m0 + lm) * N_FEAT + half * 2;  // 8B aligned
    const float* brow = tf + (n0 + lm) * N_FEAT + half * 2;

    v8f acc = {};
    for (int k = 0; k < N_FEAT; k += 4) {
        v2f a = *(const v2f*)(arow + k);
        v2f b = *(const v2f*)(brow + k);
        acc = __builtin_amdgcn_wmma_f32_16x16x4_f32(
            /*neg_a=*/false, a, /*neg_b=*/false, b,
            /*c_mod=*/(short)0, acc, /*reuse_a=*/false, /*reuse_b=*/false);
    }

    const int ncol  = n0 + lm;           // <= 79
    const int mbase = m0 + half * 8;
    #pragma unroll
    for (int r = 0; r < 8; ++r)
        E[(mbase + r) * NT80 + ncol] = acc[r];
}

// ---------------------------------------------------------------------
// Kernel 2: entropic FGW per (node, template). One wave32 per problem.
// Lane l: column j = l&7, k-group g = l>>3; 4 k-slots per lane (k = g+4i).
// ---------------------------------------------------------------------
__global__ void k_fgw(const float* __restrict__ E, const float* __restrict__ x2g,
                      const int* __restrict__ nbr,          // edge_index[1], [N_NODES*DEG]
                      const float* __restrict__ ts, const float* __restrict__ t2,
                      const float* __restrict__ c2, float* __restrict__ dist) {
    __shared__ float ts_sh[N_TPL * M_TPL * M_TPL];
    __shared__ float t2_sh[NT80];
    __shared__ float c2_sh[NT80];
    for (int i = threadIdx.x; i < N_TPL * M_TPL * M_TPL; i += blockDim.x) ts_sh[i] = ts[i];
    for (int i = threadIdx.x; i < NT80; i += blockDim.x) { t2_sh[i] = t2[i]; c2_sh[i] = c2[i]; }
    __syncthreads();

    const int wave = threadIdx.x >> 5;
    const int lane = threadIdx.x & 31;
    const int prob = blockIdx.x * 8 + wave;      // 8 waves / block
    if (prob >= N_NODES * N_TPL) return;
    const int n = prob / N_TPL;
    const int t = prob % N_TPL;

    const int j = lane & 7;      // template column
    const int g = lane >> 3;     // k group

    const float Pk = 1.0f / KLOC;
    const float Qj = 1.0f / M_TPL;
    const float tj2 = t2_sh[t * M_TPL + j];
    const float cj2 = c2_sh[t * M_TPL + j];

    float mc[4], cst[4], Km[4], T[4], u[4];
    int   valid[4];

    #pragma unroll
    for (int i = 0; i < 4; ++i) {
        const int k = g + 4 * i;
        valid[i] = (k < KLOC);
        int idx = n;
        if (valid[i] && k > 0) idx = nbr[n * DEG + (k - 1)];
        const float e  = valid[i] ? E[idx * NT80 + t * M_TPL + j] : 0.f;
        const float xx = valid[i] ? x2g[idx] : 0.f;
        mc[i]  = xx + tj2 - 2.0f * e;                        // feature cost
        cst[i] = ((k == 0) ? ((float)DEG / KLOC) : (1.0f / KLOC)) + cj2;
        T[i]   = valid[i] ? (Pk * Qj) : 0.f;                 // uniform init
        Km[i]  = 0.f; u[i] = 0.f;
    }

    float a_crs = 0.f, b_crs = 0.f;   // crs for k==0 row / k>=1 rows, column j

    for (int outer = 0; outer <= N_OUTER; ++outer) {
        // s_j = sum_{k>=1} T[k][j]   (leaf sum);  r_j = T[0][j]  (center row)
        float sp = 0.f;
        #pragma unroll
        for (int i = 0; i < 4; ++i) {
            const int k = g + 4 * i;
            if (valid[i] && k >= 1) sp += T[i];
        }
        sp += __shfl_xor(sp, 8);
        sp += __shfl_xor(sp, 16);
        const float r0 = __shfl(T[0], j);     // lanes 0..7 hold k==0 slots

        // crs = (C1 @ T) @ ts^T  -> two dot products against ts row j
        a_crs = 0.f; b_crs = 0.f;
        #pragma unroll
        for (int m = 0; m < M_TPL; ++m) {
            const float tsv = ts_sh[(t * M_TPL + j) * M_TPL + m];
            a_crs += tsv * __shfl(sp, m);
            b_crs += tsv * __shfl(r0, m);
        }
        if (outer == N_OUTER) break;          // final crs only

        // G and Gibbs kernel
        #pragma unroll
        for (int i = 0; i < 4; ++i) {
            const int k = g + 4 * i;
            const float crs = (k == 0) ? a_crs : b_crs;
            const float G = ALPHA_R * mc[i] + (1.0f - ALPHA_R) * (cst[i] - 2.0f * crs);
            Km[i] = valid[i] ? __expf(-G * (1.0f / EPS_R)) : 0.f;
        }

        // Sinkhorn projection
        float v = Qj;
        for (int s_it = 0; s_it < N_SINK; ++s_it) {
            #pragma unroll
            for (int i = 0; i < 4; ++i) {          // u_k = p / (K v)_k
                float d = Km[i] * v;
                d += __shfl_xor(d, 1);
                d += __shfl_xor(d, 2);
                d += __shfl_xor(d, 4);
                u[i] = Pk / (d + 1e-30f);
            }
            float nm = 0.f;                        // v_j = q / (K^T u)_j
            #pragma unroll
            for (int i = 0; i < 4; ++i) nm += Km[i] * u[i];
            nm += __shfl_xor(nm, 8);
            nm += __shfl_xor(nm, 16);
            v = Qj / (nm + 1e-30f);
        }
        #pragma unroll
        for (int i = 0; i < 4; ++i) T[i] = u[i] * Km[i] * v;
    }

    // dist = alpha*<Mc,T> + (1-alpha)*<const - 2*crs, T>
    float acc = 0.f;
    #pragma unroll
    for (int i = 0; i < 4; ++i) {
        const int k = g + 4 * i;
        const float crs = (k == 0) ? a_crs : b_crs;
        acc += (ALPHA_R * mc[i] + (1.0f - ALPHA_R) * (cst[i] - 2.0f * crs)) * T[i];
    }
    acc += __shfl_xor(acc, 1);
    acc += __shfl_xor(acc, 2);
    acc += __shfl_xor(acc, 4);
    acc += __shfl_xor(acc, 8);
    acc += __shfl_xor(acc, 16);
    if (lane == 0) dist[prob] = acc;
}

// ---------------------------------------------------------------------
// Kernel 3: out[n,c] = b[c] + sum_t dist[n,t] * w[c,t]
// ---------------------------------------------------------------------
__global__ void k_out(const float* __restrict__ dist, const float* __restrict__ w,
                      const float* __restrict__ b, float* __restrict__ out) {
    int i = blockIdx.x * blockDim.x + threadIdx.x;
    if (i >= N_NODES * N_CLS) return;
    const int n = i / N_CLS, c = i % N_CLS;
    float s = b[c];
    #pragma unroll
    for (int t = 0; t < N_TPL; ++t) s += dist[n * N_TPL + t] * w[c * N_TPL + t];
    out[i] = s;
}

// ---------------------------------------------------------------------
extern "C" void kernel_launch(void* const* d_in, const int* in_sizes, int n_in,
                              void* d_out, int out_size, void* d_ws, size_t ws_size,
                              hipStream_t stream) {
    (void)in_sizes; (void)n_in; (void)out_size; (void)ws_size;
    const float* x  = (const float*)d_in[0];        // [20000,128]
    const int*   ei = (const int*)d_in[1];          // [2, 240000]
    const float* tf = (const float*)d_in[2];        // [10,8,128]
    const float* ts = (const float*)d_in[3];        // [10,8,8]
    const float* w  = (const float*)d_in[4];        // [8,10]
    const float* b  = (const float*)d_in[5];        // [8]
    const int*   nbr = ei + N_NODES * DEG;          // edge_index[1]

    // workspace layout (floats)
    float* ws   = (float*)d_ws;
    float* E    = ws;                        // 20000*80  = 1,600,000
    float* X2   = E  + N_NODES * NT80;       // 20000
    float* T2   = X2 + N_NODES;              // 80
    float* C2   = T2 + NT80;                 // 80
    float* DIST = C2 + NT80;                 // 200,000

    k_tpl_pre<<<1, 256, 0, stream>>>(tf, ts, T2, C2);
    k_x2<<<(N_NODES + 255) / 256, 256, 0, stream>>>(x, X2);

    dim3 g1(N_NODES / 16, NT80 / 16);        // 1250 x 5 tiles
    k_gemm_E<<<g1, 32, 0, stream>>>(x, tf, E);

    const int nprob = N_NODES * N_TPL;       // 200,000
    k_fgw<<<nprob / 8, 256, 0, stream>>>(E, X2, nbr, ts, T2, C2, DIST);

    k_out<<<(N_NODES * N_CLS + 255) / 256, 256, 0, stream>>>(DIST, w, b, (float*)d_out);
}